// nri_encoder_32049045962801
// MI455X (gfx1250) — compile-verified
//
#include <hip/hip_runtime.h>
#include <hip/hip_bf16.h>

typedef float v2f __attribute__((ext_vector_type(2)));
typedef float v8f __attribute__((ext_vector_type(8)));

constexpr int NJ   = 58;      // joints (nodes)
constexpr int NE   = 3306;    // edges = NJ*(NJ-1)
constexpr int NH   = 512;     // hidden
constexpr int NB   = 32;      // batch
constexpr int NO   = 128;     // n_out
constexpr int NIN  = 192;     // n_in
constexpr int BJ   = NB * NJ;     // 1856 node-rows
constexpr int ROWS = NB * NE;     // 105792 edge-rows

constexpr int TM = 128, TN = 128, TK = 32;
constexpr int APAD   = 36;   // LDS stride (dwords) for A tile: conflict-free 16x4 frag reads
constexpr int BPITCH = 288;  // LDS dwords per k-pair group: 288 % 64 == 32 -> halves use disjoint banks

__device__ __forceinline__ float4 ld4(const float* p) { return *reinterpret_cast<const float4*>(p); }
__device__ __forceinline__ float2 ld2(const float* p) { return *reinterpret_cast<const float2*>(p); }

struct GemmArgs {
  const float* A; int lda;
  const float* B; int ldb;
  float* C;       int ldc;
  int M, N, K;
  // MODE 1 (fused edge MLP2 skip-GEMM) extras:
  const float* hs;    const float* hd;   const float* abias;   // regenerate e_raw on the fly
  const float* h2s;   const float* h2d;  const float* ebias;   // epilogue gather + folded bias
  const int* srcIdx;  const int* dstIdx;
  float* psum;        float* psq;                              // deterministic BN partials
};

// MODE 0: C = A @ B
// MODE 1: A regenerated as relu(hs[src]+hd[dst]+abias); epilogue: relu(acc + h2s[src]+h2d[dst]+ebias)
//         -> store C (e2) + per-block BN partial sums (ordered, deterministic)
// MODE 2: C = A @ B + ebias[col]
template<int MODE>
__launch_bounds__(256)
__global__ void gemm_wmma(GemmArgs g) {
  __shared__ float As[TM * APAD];                 // 18 KB
  __shared__ float Bs[(TK / 2) * BPITCH];        // 18 KB, k-pair interleaved
  constexpr int XI = (MODE == 1) ? 2 * TM : 1;
  constexpr int XF = (MODE == 1) ? 2 * 16 * TN : 1;
  __shared__ int   gsrow[XI];
  __shared__ float stats[XF];

  const int tid   = threadIdx.x;
  const int w     = tid >> 5;
  const int lane  = tid & 31;
  const int mrow  = lane & 15;
  const int khalf = lane >> 4;
  const int m0 = blockIdx.y * TM;
  const int n0 = blockIdx.x * TN;

  if constexpr (MODE == 1) {
    if (tid < TM) {
      int R = m0 + tid;
      int b = 0, e = 0;
      if (R < g.M) { b = R / NE; e = R - b * NE; }
      gsrow[tid]      = (b * NJ + g.srcIdx[e]) * NH;
      gsrow[TM + tid] = (b * NJ + g.dstIdx[e]) * NH;
    }
    __syncthreads();
  }

  v8f acc[8] = {};

  const int arow0 = tid >> 3;        // 0..31 (+p*32)
  const int acol  = (tid & 7) * 4;   // 0..28
  const int bcol  = lane * 4;        // 0..124
  const int bpr0  = tid >> 5;        // 0..7 (+p*8) k-pair index
  const int nkc   = g.K / TK;

  // fragment base pointers (fixed across K-chunks; LDS tiles are reused in place)
  const float* aRow  = &As[(w * 16 + mrow) * APAD + 2 * khalf];
  const float* bBase = &Bs[khalf * BPITCH + 2 * mrow];

  for (int kc = 0; kc < nkc; ++kc) {
    const int k0 = kc * TK;
    #pragma unroll
    for (int p = 0; p < 4; ++p) {           // A tile: 128 x 32
      int r = arow0 + p * 32;
      float4 v;
      if constexpr (MODE == 1) {
        float4 a = ld4(g.hs + gsrow[r] + k0 + acol);
        float4 d = ld4(g.hd + gsrow[TM + r] + k0 + acol);
        float4 bb = ld4(g.abias + k0 + acol);
        v.x = fmaxf(a.x + d.x + bb.x, 0.f);
        v.y = fmaxf(a.y + d.y + bb.y, 0.f);
        v.z = fmaxf(a.z + d.z + bb.z, 0.f);
        v.w = fmaxf(a.w + d.w + bb.w, 0.f);
      } else {
        int R = m0 + r;
        if (R < g.M) v = ld4(g.A + (size_t)R * g.lda + k0 + acol);
        else         v = make_float4(0.f, 0.f, 0.f, 0.f);
      }
      *reinterpret_cast<float4*>(&As[r * APAD + acol]) = v;
    }
    #pragma unroll
    for (int p = 0; p < 2; ++p) {           // B tile: 32 x 128, k-pair interleaved
      int pr = bpr0 + p * 8;                // 0..15
      const float* b0 = g.B + (size_t)(k0 + 2 * pr) * g.ldb + n0 + bcol;
      float4 va = ld4(b0);
      float4 vb = ld4(b0 + g.ldb);
      float* o = &Bs[pr * BPITCH + 2 * bcol];
      v2f t0; t0.x = va.x; t0.y = vb.x;
      v2f t1; t1.x = va.y; t1.y = vb.y;
      v2f t2; t2.x = va.z; t2.y = vb.z;
      v2f t3; t3.x = va.w; t3.y = vb.w;
      reinterpret_cast<v2f*>(o)[0] = t0;
      reinterpret_cast<v2f*>(o)[1] = t1;
      reinterpret_cast<v2f*>(o)[2] = t2;
      reinterpret_cast<v2f*>(o)[3] = t3;
    }
    __syncthreads();

    // double-buffered fragment pipeline over 8 k-steps
    v2f a_cur = *reinterpret_cast<const v2f*>(aRow);
    v2f b_cur[8];
    #pragma unroll
    for (int t = 0; t < 8; ++t)
      b_cur[t] = *reinterpret_cast<const v2f*>(bBase + 32 * t);

    #pragma unroll
    for (int kk = 0; kk < 8; ++kk) {
      v2f a_nxt;
      v2f b_nxt[8];
      if (kk < 7) {
        a_nxt = *reinterpret_cast<const v2f*>(aRow + (kk + 1) * 4);
        #pragma unroll
        for (int t = 0; t < 8; ++t)
          b_nxt[t] = *reinterpret_cast<const v2f*>(bBase + (kk + 1) * 2 * BPITCH + 32 * t);
      }
      #pragma unroll
      for (int t = 0; t < 8; ++t)
        acc[t] = __builtin_amdgcn_wmma_f32_16x16x4_f32(
            false, a_cur, false, b_cur[t], (short)0, acc[t], false, false);
      if (kk < 7) {
        a_cur = a_nxt;
        #pragma unroll
        for (int t = 0; t < 8; ++t) b_cur[t] = b_nxt[t];
      }
    }
    __syncthreads();
  }

  const int Rbase = m0 + w * 16 + khalf * 8;
  if constexpr (MODE == 0) {
    #pragma unroll
    for (int t = 0; t < 8; ++t) {
      int Cg = n0 + t * 16 + mrow;
      #pragma unroll
      for (int r = 0; r < 8; ++r) {
        int R = Rbase + r;
        if (R < g.M) g.C[(size_t)R * g.ldc + Cg] = acc[t][r];
      }
    }
  } else if constexpr (MODE == 2) {
    #pragma unroll
    for (int t = 0; t < 8; ++t) {
      int Cg = n0 + t * 16 + mrow;
      float eb = g.ebias[Cg];
      #pragma unroll
      for (int r = 0; r < 8; ++r) {
        int R = Rbase + r;
        if (R < g.M) g.C[(size_t)R * g.ldc + Cg] = acc[t][r] + eb;
      }
    }
  } else if constexpr (MODE == 1) {
    const int lr0 = w * 16 + khalf * 8;
    float* lss = stats;
    float* lsq = stats + 16 * TN;
    #pragma unroll
    for (int t = 0; t < 8; ++t) {
      int Cl = t * 16 + mrow;
      int Cg = n0 + Cl;
      float eb = g.ebias[Cg];
      float s = 0.f, q = 0.f;
      #pragma unroll
      for (int r = 0; r < 8; ++r) {
        int R = Rbase + r;
        int lr = lr0 + r;
        float val = acc[t][r] + g.h2s[gsrow[lr] + Cg] + g.h2d[gsrow[TM + lr] + Cg] + eb;
        val = fmaxf(val, 0.f);
        if (R < g.M) {
          g.C[(size_t)R * g.ldc + Cg] = val;
          s += val; q += val * val;
        }
      }
      int slot = w * 2 + khalf;
      lss[slot * TN + Cl] = s;
      lsq[slot * TN + Cl] = q;
    }
    __syncthreads();
    if (tid < TN) {  // ordered 16-slot reduction -> deterministic partials
      float S = 0.f, Q = 0.f;
      #pragma unroll
      for (int slot = 0; slot < 16; ++slot) { S += lss[slot * TN + tid]; Q += lsq[slot * TN + tid]; }
      g.psum[(size_t)blockIdx.y * NH + n0 + tid] = S;
      g.psq [(size_t)blockIdx.y * NH + n0 + tid] = Q;
    }
  }
}

// --- graph prep: normalized adjacency + CSR(by dst), deterministic -----------
__global__ void build_graph_kernel(const int* ei, float* Aout, int* csrOff, int* csrSrc) {
  const int* src = ei;
  const int* dst = ei + NE;
  __shared__ int   adjL[NJ * NJ];
  __shared__ int   degL[NJ];
  __shared__ float dinv[NJ];
  __shared__ int   offL[NJ + 1];
  int tid = threadIdx.x;
  for (int i = tid; i < NJ * NJ; i += 256) adjL[i] = 0;
  if (tid < NJ) degL[tid] = 0;
  __syncthreads();
  for (int e = tid; e < NE; e += 256) {
    atomicAdd(&adjL[dst[e] * NJ + src[e]], 1);
    atomicAdd(&degL[dst[e]], 1);
  }
  __syncthreads();
  if (tid == 0) {  // serial CSR fill: deterministic edge order per dst
    int off = 0;
    for (int n = 0; n < NJ; ++n) { offL[n] = off; off += degL[n]; }
    offL[NJ] = off;
    int cur[NJ];
    for (int n = 0; n < NJ; ++n) cur[n] = offL[n];
    for (int e = 0; e < NE; ++e) csrSrc[cur[dst[e]]++] = src[e];
  }
  __syncthreads();
  if (tid <= NJ) csrOff[tid] = offL[tid];
  if (tid < NJ) {
    int rs = 1;  // + self loop
    for (int j = 0; j < NJ; ++j) rs += adjL[tid * NJ + j];
    dinv[tid] = rsqrtf((float)rs);
  }
  __syncthreads();
  for (int i = tid; i < NJ * NJ; i += 256) {
    int r = i / NJ, c = i % NJ;
    float v = (float)adjL[i] + (r == c ? 1.f : 0.f);
    Aout[i] = dinv[r] * v * dinv[c];
  }
}

// h = relu(A @ t + bias), one block per (b, node), 2 channels/thread
__global__ void gcn_apply_kernel(const float* Anorm, const float* tin, const float* bias, float* hout) {
  int bi = blockIdx.x;
  int b = bi / NJ, i = bi % NJ;
  __shared__ float Arow[NJ];
  int tid = threadIdx.x;
  if (tid < NJ) Arow[tid] = Anorm[i * NJ + tid];
  __syncthreads();
  int c = tid * 2;
  float ax = bias[c], ay = bias[c + 1];
  const float* base = tin + (size_t)b * NJ * NH + c;
  for (int j = 0; j < NJ; ++j) {
    float a = Arow[j];
    float2 tv = ld2(base + j * NH);
    ax += a * tv.x; ay += a * tv.y;
  }
  float* o = hout + (size_t)bi * NH + c;
  o[0] = fmaxf(ax, 0.f);
  o[1] = fmaxf(ay, 0.f);
}

// BN1 stats: e = relu(hs[src]+hd[dst]+b) computed on the fly, per-channel partials
__global__ void edge1_stats_kernel(const int* ei, const float* hs, const float* hd,
                                   const float* mlp1b, float* psum, float* psq) {
  const int* src = ei;
  const int* dst = ei + NE;
  __shared__ int gs[64], gd[64];
  int tid = threadIdx.x;
  int r0 = blockIdx.x * 64;
  if (tid < 64) {
    int R = r0 + tid;
    int b = R / NE, e = R - b * NE;
    gs[tid] = (b * NJ + src[e]) * NH;
    gd[tid] = (b * NJ + dst[e]) * NH;
  }
  __syncthreads();
  int c = tid * 2;
  float2 bb = ld2(mlp1b + c);
  float s0 = 0, s1v = 0, q0 = 0, q1 = 0;
  for (int r = 0; r < 64; ++r) {
    float2 a = ld2(hs + gs[r] + c);
    float2 d = ld2(hd + gd[r] + c);
    float v0 = fmaxf(a.x + d.x + bb.x, 0.f);
    float v1 = fmaxf(a.y + d.y + bb.y, 0.f);
    s0 += v0; s1v += v1; q0 += v0 * v0; q1 += v1 * v1;
  }
  size_t o = (size_t)blockIdx.x * NH + c;
  psum[o] = s0; psum[o + 1] = s1v;
  psq[o]  = q0; psq[o + 1]  = q1;
}

// finalize BN: s = g*rsqrt(var+eps), c = beta - mu*s   (1 block, 512 threads)
__global__ void bn_finalize_kernel(const float* psum, const float* psq, int nparts,
                                   const float* gamma, const float* beta, float count,
                                   float* sOut, float* cOut) {
  int ch = threadIdx.x;
  float S = 0, Q = 0;
  for (int p = 0; p < nparts; ++p) { S += psum[(size_t)p * NH + ch]; Q += psq[(size_t)p * NH + ch]; }
  float mu = S / count;
  float var = Q / count - mu * mu;
  float sc = gamma[ch] * rsqrtf(var + 1e-5f);
  sOut[ch] = sc;
  cOut[ch] = beta[ch] - mu * sc;
}

__global__ void scale_rows_kernel(const float* Win, const float* s, float* Wout, int N) {
  int idx = blockIdx.x * 256 + threadIdx.x;
  if (idx >= NH * N) return;
  Wout[idx] = s[idx / N] * Win[idx];
}

// bout[n] = b0[n] + sum_ch cvec[ch] * Win[ch*N+n]   (1 block, N threads)
__global__ void fold_bias_kernel(const float* Win, const float* cvec, const float* b0,
                                 float* bout, int N) {
  int n = threadIdx.x;
  if (n >= N) return;
  float a = b0[n];
  for (int ch = 0; ch < NH; ++ch) a += cvec[ch] * Win[(size_t)ch * N + n];
  bout[n] = a;
}

// node[b,n] = s1 * sum_{e: dst=n} relu(hs[b,src]+hd[b,n]+b1) + deg*c1   (CSR, deterministic)
__global__ void node_gather_kernel(const float* hs, const float* hd, const float* mlp1b,
                                   const int* csrOff, const int* csrSrc,
                                   const float* s1, const float* c1, float* node) {
  int bi = blockIdx.x;
  int b = bi / NJ, n = bi % NJ;
  __shared__ int snl[64];
  int tid = threadIdx.x;
  int off = csrOff[n];
  int deg = csrOff[n + 1] - off;
  if (deg > 64) deg = 64;
  if (tid < deg) snl[tid] = csrSrc[off + tid];
  __syncthreads();
  int c = tid * 2;
  float2 hdc = ld2(hd + (size_t)bi * NH + c);
  float2 bb = ld2(mlp1b + c);
  hdc.x += bb.x; hdc.y += bb.y;
  float a0 = 0, a1 = 0;
  for (int j = 0; j < deg; ++j) {
    float2 v = ld2(hs + (size_t)(b * NJ + snl[j]) * NH + c);
    a0 += fmaxf(v.x + hdc.x, 0.f);
    a1 += fmaxf(v.y + hdc.y, 0.f);
  }
  float dg = (float)deg;
  size_t o = (size_t)bi * NH + c;
  node[o]     = s1[c] * a0     + dg * c1[c];
  node[o + 1] = s1[c + 1] * a1 + dg * c1[c + 1];
}

extern "C" void kernel_launch(void* const* d_in, const int* in_sizes, int n_in,
                              void* d_out, int out_size, void* d_ws, size_t ws_size,
                              hipStream_t stream) {
  (void)in_sizes; (void)n_in; (void)out_size; (void)ws_size;
  const float* x      = (const float*)d_in[0];
  const int*   ei     = (const int*)  d_in[1];
  const float* gcn1_w = (const float*)d_in[2];
  const float* gcn1_b = (const float*)d_in[3];
  const float* mlp1_w = (const float*)d_in[4];
  const float* mlp1_b = (const float*)d_in[5];
  const float* bn1_g  = (const float*)d_in[6];
  const float* bn1_b  = (const float*)d_in[7];
  const float* gcn2_w = (const float*)d_in[8];
  const float* gcn2_b = (const float*)d_in[9];
  const float* mlp2_w = (const float*)d_in[10];
  const float* mlp2_b = (const float*)d_in[11];
  const float* bn2_g  = (const float*)d_in[12];
  const float* bn2_b  = (const float*)d_in[13];
  const float* fc_w   = (const float*)d_in[14];
  const float* fc_b   = (const float*)d_in[15];
  float* out = (float*)d_out;

  char* p = (char*)d_ws;
  auto alloc = [&](size_t bytes) { char* r = p; p += (bytes + 255) & ~size_t(255); return r; };

  const int NPART1 = ROWS / 64;              // 1653
  const int NPART2 = (ROWS + TM - 1) / TM;   // 827

  float* Anorm = (float*)alloc((size_t)NJ * NJ * 4);
  int* csrOff  = (int*)alloc((NJ + 1) * 4);
  int* csrSrc  = (int*)alloc((size_t)NE * 4);
  float* yb    = (float*)alloc((size_t)BJ * NH * 4);
  float* hb    = (float*)alloc((size_t)BJ * NH * 4);
  float* hsB   = (float*)alloc((size_t)BJ * NH * 4);
  float* hdB   = (float*)alloc((size_t)BJ * NH * 4);
  float* nodeB = (float*)alloc((size_t)BJ * NH * 4);
  float* tb    = (float*)alloc((size_t)BJ * NH * 4);
  float* h2b   = (float*)alloc((size_t)BJ * NH * 4);
  float* h2sB  = (float*)alloc((size_t)BJ * NH * 4);
  float* h2dB  = (float*)alloc((size_t)BJ * NH * 4);
  float* e2    = (float*)alloc((size_t)ROWS * NH * 4);      // 217 MB
  float* Wskip = (float*)alloc((size_t)NH * NH * 4);
  float* Wfc   = (float*)alloc((size_t)NH * NO * 4);
  float* mlp2bAdj = (float*)alloc(NH * 4);
  float* biasOut  = (float*)alloc(NO * 4);
  float* s1 = (float*)alloc(NH * 4);
  float* c1 = (float*)alloc(NH * 4);
  float* s2 = (float*)alloc(NH * 4);
  float* c2 = (float*)alloc(NH * 4);
  float* p1s = (float*)alloc((size_t)NPART1 * NH * 4);
  float* p1q = (float*)alloc((size_t)NPART1 * NH * 4);
  float* p2s = (float*)alloc((size_t)NPART2 * NH * 4);
  float* p2q = (float*)alloc((size_t)NPART2 * NH * 4);

  const dim3 gSmall(NH / TN, (BJ + TM - 1) / TM);   // (4, 15)

  // 1) graph prep
  build_graph_kernel<<<1, 256, 0, stream>>>(ei, Anorm, csrOff, csrSrc);

  // 2) y = x @ gcn1_w ; h = relu(A @ y + b1)
  GemmArgs a1{}; a1.A = x; a1.lda = NIN; a1.B = gcn1_w; a1.ldb = NH; a1.C = yb; a1.ldc = NH;
  a1.M = BJ; a1.N = NH; a1.K = NIN;
  gemm_wmma<0><<<gSmall, 256, 0, stream>>>(a1);
  gcn_apply_kernel<<<BJ, 256, 0, stream>>>(Anorm, yb, gcn1_b, hb);

  // 3) hs = h @ W1_top ; hd = h @ W1_bot
  GemmArgs a2{}; a2.A = hb; a2.lda = NH; a2.ldb = NH; a2.ldc = NH; a2.M = BJ; a2.N = NH; a2.K = NH;
  a2.B = mlp1_w;            a2.C = hsB; gemm_wmma<0><<<gSmall, 256, 0, stream>>>(a2);
  a2.B = mlp1_w + NH * NH;  a2.C = hdB; gemm_wmma<0><<<gSmall, 256, 0, stream>>>(a2);

  // 4) BN1 stats (e computed on the fly) + fold BN1 into skip weights/bias
  edge1_stats_kernel<<<NPART1, 256, 0, stream>>>(ei, hsB, hdB, mlp1_b, p1s, p1q);
  bn_finalize_kernel<<<1, NH, 0, stream>>>(p1s, p1q, NPART1, bn1_g, bn1_b, (float)ROWS, s1, c1);
  scale_rows_kernel<<<(NH * NH + 255) / 256, 256, 0, stream>>>(mlp2_w + 2 * NH * NH, s1, Wskip, NH);
  fold_bias_kernel<<<1, NH, 0, stream>>>(mlp2_w + 2 * NH * NH, c1, mlp2_b, mlp2bAdj, NH);

  // 5) node = scatter-add of BN1(e) over dst (folded CSR gather)
  node_gather_kernel<<<BJ, 256, 0, stream>>>(hsB, hdB, mlp1_b, csrOff, csrSrc, s1, c1, nodeB);

  // 6) t = node @ gcn2_w ; h2 = relu(A @ t + b2)
  GemmArgs a3{}; a3.A = nodeB; a3.lda = NH; a3.B = gcn2_w; a3.ldb = NH; a3.C = tb; a3.ldc = NH;
  a3.M = BJ; a3.N = NH; a3.K = NH;
  gemm_wmma<0><<<gSmall, 256, 0, stream>>>(a3);
  gcn_apply_kernel<<<BJ, 256, 0, stream>>>(Anorm, tb, gcn2_b, h2b);

  // 7) h2s = h2 @ W2_top ; h2d = h2 @ W2_mid
  GemmArgs a4{}; a4.A = h2b; a4.lda = NH; a4.ldb = NH; a4.ldc = NH; a4.M = BJ; a4.N = NH; a4.K = NH;
  a4.B = mlp2_w;           a4.C = h2sB; gemm_wmma<0><<<gSmall, 256, 0, stream>>>(a4);
  a4.B = mlp2_w + NH * NH; a4.C = h2dB; gemm_wmma<0><<<gSmall, 256, 0, stream>>>(a4);

  // 8) BIG fused GEMM: e2 = relu(e_raw @ Wskip' + h2s[src] + h2d[dst] + b') + BN2 partials
  GemmArgs a5{}; a5.B = Wskip; a5.ldb = NH; a5.C = e2; a5.ldc = NH;
  a5.M = ROWS; a5.N = NH; a5.K = NH;
  a5.hs = hsB; a5.hd = hdB; a5.abias = mlp1_b;
  a5.h2s = h2sB; a5.h2d = h2dB; a5.ebias = mlp2bAdj;
  a5.srcIdx = ei; a5.dstIdx = ei + NE;
  a5.psum = p2s; a5.psq = p2q;
  gemm_wmma<1><<<dim3(NH / TN, NPART2), 256, 0, stream>>>(a5);

  // 9) BN2 fold into fc
  bn_finalize_kernel<<<1, NH, 0, stream>>>(p2s, p2q, NPART2, bn2_g, bn2_b, (float)ROWS, s2, c2);
  scale_rows_kernel<<<(NH * NO + 255) / 256, 256, 0, stream>>>(fc_w, s2, Wfc, NO);
  fold_bias_kernel<<<1, NO, 0, stream>>>(fc_w, c2, fc_b, biasOut, NO);

  // 10) out = e2 @ Wfc' + biasOut
  GemmArgs a6{}; a6.A = e2; a6.lda = NH; a6.B = Wfc; a6.ldb = NO; a6.C = out; a6.ldc = NO;
  a6.M = ROWS; a6.N = NO; a6.K = NH; a6.ebias = biasOut;
  gemm_wmma<2><<<dim3(1, NPART2), 256, 0, stream>>>(a6);
}